// HeteroEncoder_61864708931626
// MI455X (gfx1250) — compile-verified
//
#include <hip/hip_runtime.h>
#include <hip/hip_bf16.h>

// ---------------------------------------------------------------------------
// Hetero GraphSAGE (2 layers) for MI455X / gfx1250.
//   - segment_sum via wave-per-edge global_atomic_add_f32 scatter (L2-local)
//   - neighbor counts computed once (graph is static across both layers)
//   - dense 128x128 linears via V_WMMA_F32_16X16X4_F32 (full f32 precision),
//     with A/x tiles staged in padded LDS (conflict-free ds_load_b64 fragments)
// ---------------------------------------------------------------------------

#define N_GENE 100000
#define N_DIS  30000
#define E_GG   1600000
#define E_GD   1000000
#define E_DG   1000000
#define DIMF   128          // feature dim for every layer (D == H == O == 128)
#define LDSW   132          // padded LDS row stride (132 mod 64 = 4 -> no bank conflicts)
#define TILE_F (16 * LDSW)  // floats per staged 16x128 tile

typedef __attribute__((ext_vector_type(2))) float v2f;
typedef __attribute__((ext_vector_type(8))) float v8f;

// ---------------------------------------------------------------------------
// f32 WMMA: D = A(16x4) * B(4x16) + C(16x16), wave32.
// ---------------------------------------------------------------------------
__device__ __forceinline__ v8f wmma_f32(v2f a, v2f b, v8f c) {
    return __builtin_amdgcn_wmma_f32_16x16x4_f32(false, a, false, b,
                                                 (short)0, c, false, false);
}

// ---------------------------------------------------------------------------
// Zero-fill (float4 granularity, grid-stride)
// ---------------------------------------------------------------------------
__global__ void zero_f4_kernel(float4* __restrict__ p, long n4) {
    long i      = (long)blockIdx.x * blockDim.x + threadIdx.x;
    long stride = (long)gridDim.x * blockDim.x;
    float4 z; z.x = 0.f; z.y = 0.f; z.z = 0.f; z.w = 0.f;
    for (; i < n4; i += stride) p[i] = z;
}

// ---------------------------------------------------------------------------
// Scatter: one wave per edge. Lane l accumulates features [4l, 4l+4) of the
// source row into agg[dst]. COUNT: lane 0 bumps the neighbor count (layer 1
// only — the graph, and therefore the counts, are identical in layer 2).
// ---------------------------------------------------------------------------
template <bool COUNT>
__global__ void sage_scatter_kernel(const float* __restrict__ xsrc,
                                    const int*  __restrict__ src,
                                    const int*  __restrict__ dst,
                                    float* __restrict__ agg,
                                    float* __restrict__ cnt,
                                    int n_edges) {
    int wid  = (int)((blockIdx.x * (long)blockDim.x + threadIdx.x) >> 5);
    int lane = threadIdx.x & 31;
    if (wid >= n_edges) return;
    int s = src[wid];
    int d = dst[wid];
    const float4 v = ((const float4*)(xsrc + (long)s * DIMF))[lane];
    float* ap = agg + (long)d * DIMF + lane * 4;
    atomicAdd(ap + 0, v.x);
    atomicAdd(ap + 1, v.y);
    atomicAdd(ap + 2, v.z);
    atomicAdd(ap + 3, v.w);
    if (COUNT && lane == 0) atomicAdd(cnt + d, 1.0f);
}

// ---------------------------------------------------------------------------
// Stage a 16x128 f32 tile (rows row0..row0+15 of a [n,128] matrix) into LDS
// with row stride LDSW, optionally scaling each row by 1/max(cnt[row],1).
// 256 threads, fully coalesced float4 global loads (2 per thread).
// ---------------------------------------------------------------------------
__device__ __forceinline__ void stage_tile(float* __restrict__ sm,
                                           const float* __restrict__ g,
                                           const float* __restrict__ cnt,
                                           int row0, int tid) {
    #pragma unroll
    for (int e = tid; e < 512; e += 256) {          // 512 float4 = 16x128 floats
        const int row = e >> 5;                      // 0..15
        const int c4  = e & 31;                      // float4 column
        float4 v = ((const float4*)g)[(long)(row0 + row) * 32 + c4];
        if (cnt) {
            const float s = 1.0f / fmaxf(cnt[row0 + row], 1.0f);
            v.x *= s; v.y *= s; v.z *= s; v.w *= s;
        }
        float2* d = (float2*)&sm[row * LDSW + c4 * 4];
        d[0] = make_float2(v.x, v.y);
        d[1] = make_float2(v.z, v.w);
    }
}

// ---------------------------------------------------------------------------
// Fused SAGE combine for a 16-row tile of destination nodes:
//   out = [relu]( (aggA/cntA)@WlA + bA + x@WrA  [ + (aggB/cntB)@WlB + bB + x@WrB ] )
// Block = 256 threads = 8 waves; wave w owns output columns [16w, 16w+16).
//
// Fragment layout (wave32, f32 WMMA):
//   A 16x4 : lanes 0-15 -> row M=lane, {K, K+1}; lanes 16-31 -> row M=lane-16, {K+2, K+3}
//   B 4x16 : lanes 0-15 -> col N=lane, rows {K, K+1}; lanes 16-31 -> col N=lane-16, {K+2, K+3}
//   C 16x16: vgpr r, lane L -> row = r + (L>=16 ? 8 : 0), col = L%16
// A fragments come from padded LDS (ds_load_b64, conflict-free); B fragments
// are coalesced global loads of the weight matrices (L2-resident, 64 KB each).
// ---------------------------------------------------------------------------
template <bool HAS_B, bool RELU>
__global__ void sage_combine_kernel(float* __restrict__ out,
                                    const float* __restrict__ aggA,
                                    const float* __restrict__ cntA,
                                    const float* __restrict__ WlA,
                                    const float* __restrict__ bA,
                                    const float* __restrict__ x,
                                    const float* __restrict__ WrA,
                                    const float* __restrict__ aggB,
                                    const float* __restrict__ cntB,
                                    const float* __restrict__ WlB,
                                    const float* __restrict__ bB,
                                    const float* __restrict__ WrB) {
    __shared__ float sm[(HAS_B ? 3 : 2) * TILE_F];
    float* smA = sm;                 // mean_A tile (pre-scaled)
    float* smX = sm + TILE_F;        // root x tile
    float* smB = HAS_B ? (sm + 2 * TILE_F) : nullptr;

    const int tid  = threadIdx.x;
    const int lane = tid & 31;
    const int wave = tid >> 5;
    const int hi   = lane >> 4;
    const int l15  = lane & 15;

    const int row0 = blockIdx.x * 16;
    const int ncol = wave * 16 + l15;

    // ---- stage tiles (coalesced) ----
    stage_tile(smA, aggA, cntA, row0, tid);
    stage_tile(smX, x,    nullptr, row0, tid);
    if (HAS_B) stage_tile(smB, aggB, cntB, row0, tid);
    __syncthreads();

    // LDS fragment base for this lane's A-row (float2 view; 8B aligned)
    const float2* la = (const float2*)&smA[l15 * LDSW];
    const float2* lx = (const float2*)&smX[l15 * LDSW];
    const float2* lb = HAS_B ? (const float2*)&smB[l15 * LDSW] : nullptr;

    v8f c = {0.f, 0.f, 0.f, 0.f, 0.f, 0.f, 0.f, 0.f};

    #pragma unroll 4
    for (int ko = 0; ko < DIMF; ko += 4) {
        const int ka = ko + 2 * hi;            // this lane's K pair {ka, ka+1}
        const int kh = ka >> 1;

        // ---- mean_A @ WlA ----
        float2 am = la[kh];
        v2f a;  a.x = am.x;  a.y = am.y;
        v2f b;  b.x = WlA[(long)ka * DIMF + ncol];
                b.y = WlA[(long)(ka + 1) * DIMF + ncol];
        c = wmma_f32(a, b, c);

        // ---- x @ WrA (root) ----
        float2 xm = lx[kh];
        v2f ax; ax.x = xm.x; ax.y = xm.y;
        v2f br; br.x = WrA[(long)ka * DIMF + ncol];
                br.y = WrA[(long)(ka + 1) * DIMF + ncol];
        c = wmma_f32(ax, br, c);

        if (HAS_B) {
            // ---- mean_B @ WlB ----
            float2 bm = lb[kh];
            v2f a2; a2.x = bm.x; a2.y = bm.y;
            v2f b2; b2.x = WlB[(long)ka * DIMF + ncol];
                    b2.y = WlB[(long)(ka + 1) * DIMF + ncol];
            c = wmma_f32(a2, b2, c);

            // ---- x @ WrB (root, second edge type) ----
            v2f b3; b3.x = WrB[(long)ka * DIMF + ncol];
                    b3.y = WrB[(long)(ka + 1) * DIMF + ncol];
            c = wmma_f32(ax, b3, c);
        }
    }

    // Bias (per output column; identical for all 8 accumulator rows).
    float bias = bA[ncol];
    if (HAS_B) bias += bB[ncol];

    #pragma unroll
    for (int r = 0; r < 8; ++r) {
        float v = c[r] + bias;
        if (RELU) v = fmaxf(v, 0.0f);
        const int orow = row0 + r + hi * 8;
        out[(long)orow * DIMF + ncol] = v;
    }
}

// ---------------------------------------------------------------------------
// Host-side orchestration
// ---------------------------------------------------------------------------
extern "C" void kernel_launch(void* const* d_in, const int* in_sizes, int n_in,
                              void* d_out, int out_size, void* d_ws, size_t ws_size,
                              hipStream_t stream) {
    // ---- inputs (setup_inputs order) ----
    const float* x_gene    = (const float*)d_in[0];
    const float* x_disease = (const float*)d_in[1];
    const int* e_gg_src = (const int*)d_in[2];
    const int* e_gg_dst = (const int*)d_in[3];
    const int* e_gd_src = (const int*)d_in[4];
    const int* e_gd_dst = (const int*)d_in[5];
    const int* e_dg_src = (const int*)d_in[6];
    const int* e_dg_dst = (const int*)d_in[7];
    const float* W1_gg_l = (const float*)d_in[8];
    const float* b1_gg   = (const float*)d_in[9];
    const float* W1_gg_r = (const float*)d_in[10];
    const float* W1_gd_l = (const float*)d_in[11];
    const float* b1_gd   = (const float*)d_in[12];
    const float* W1_gd_r = (const float*)d_in[13];
    const float* W1_dg_l = (const float*)d_in[14];
    const float* b1_dg   = (const float*)d_in[15];
    const float* W1_dg_r = (const float*)d_in[16];
    const float* W2_gg_l = (const float*)d_in[17];
    const float* b2_gg   = (const float*)d_in[18];
    const float* W2_gg_r = (const float*)d_in[19];
    const float* W2_gd_l = (const float*)d_in[20];
    const float* b2_gd   = (const float*)d_in[21];
    const float* W2_gd_r = (const float*)d_in[22];
    const float* W2_dg_l = (const float*)d_in[23];
    const float* b2_dg   = (const float*)d_in[24];
    const float* W2_dg_r = (const float*)d_in[25];

    // ---- workspace layout (floats) ----
    float* ws = (float*)d_ws;
    const long SZ_G = (long)N_GENE * DIMF;     // 12.8M
    const long SZ_D = (long)N_DIS  * DIMF;     //  3.84M
    float* agg_g_a = ws;                       // gg -> gene
    float* agg_g_b = agg_g_a + SZ_G;           // dg -> gene
    float* agg_d   = agg_g_b + SZ_G;           // gd -> disease
    float* cnt_g_a = agg_d   + SZ_D;           // N_GENE
    float* cnt_g_b = cnt_g_a + N_GENE;         // N_GENE
    float* cnt_d   = cnt_g_b + N_GENE;         // N_DIS
    float* h_gene  = cnt_d   + N_DIS;          // SZ_G
    float* h_dis   = h_gene  + SZ_G;           // SZ_D

    const long agg_floats = 2 * SZ_G + SZ_D;                 // 29,440,000 (/4 ok)
    const long all_floats = agg_floats + 2 * N_GENE + N_DIS; // + counts (/4 ok)

    float* o_gene = (float*)d_out;
    float* o_dis  = o_gene + SZ_G;

    const int ZB = 2048;
    const int SCAT_TPB = 256;            // 8 edges (waves) per block
    dim3 comb_block(256);
    dim3 grid_gene(N_GENE / 16);         // 6250, exact
    dim3 grid_dis(N_DIS / 16);           // 1875, exact

    // ======================= LAYER 1 =======================
    hipLaunchKernelGGL(zero_f4_kernel, dim3(ZB), dim3(256), 0, stream,
                       (float4*)ws, all_floats / 4);

    hipLaunchKernelGGL((sage_scatter_kernel<true>), dim3((E_GG + 7) / 8), dim3(SCAT_TPB), 0, stream,
                       x_gene, e_gg_src, e_gg_dst, agg_g_a, cnt_g_a, E_GG);
    hipLaunchKernelGGL((sage_scatter_kernel<true>), dim3((E_DG + 7) / 8), dim3(SCAT_TPB), 0, stream,
                       x_disease, e_dg_src, e_dg_dst, agg_g_b, cnt_g_b, E_DG);
    hipLaunchKernelGGL((sage_scatter_kernel<true>), dim3((E_GD + 7) / 8), dim3(SCAT_TPB), 0, stream,
                       x_gene, e_gd_src, e_gd_dst, agg_d, cnt_d, E_GD);

    // h_gene = relu(sage_gg(x_gene) + sage_dg(x_disease->gene))
    hipLaunchKernelGGL((sage_combine_kernel<true, true>), grid_gene, comb_block, 0, stream,
                       h_gene,
                       agg_g_a, cnt_g_a, W1_gg_l, b1_gg, x_gene, W1_gg_r,
                       agg_g_b, cnt_g_b, W1_dg_l, b1_dg, W1_dg_r);
    // h_dis = relu(sage_gd(x_gene->disease))
    hipLaunchKernelGGL((sage_combine_kernel<false, true>), grid_dis, comb_block, 0, stream,
                       h_dis,
                       agg_d, cnt_d, W1_gd_l, b1_gd, x_disease, W1_gd_r,
                       nullptr, nullptr, nullptr, nullptr, nullptr);

    // ======================= LAYER 2 =======================
    // Counts are graph-static: zero only the agg region, keep cnt_* buffers.
    hipLaunchKernelGGL(zero_f4_kernel, dim3(ZB), dim3(256), 0, stream,
                       (float4*)ws, agg_floats / 4);

    hipLaunchKernelGGL((sage_scatter_kernel<false>), dim3((E_GG + 7) / 8), dim3(SCAT_TPB), 0, stream,
                       h_gene, e_gg_src, e_gg_dst, agg_g_a, cnt_g_a, E_GG);
    hipLaunchKernelGGL((sage_scatter_kernel<false>), dim3((E_DG + 7) / 8), dim3(SCAT_TPB), 0, stream,
                       h_dis, e_dg_src, e_dg_dst, agg_g_b, cnt_g_b, E_DG);
    hipLaunchKernelGGL((sage_scatter_kernel<false>), dim3((E_GD + 7) / 8), dim3(SCAT_TPB), 0, stream,
                       h_gene, e_gd_src, e_gd_dst, agg_d, cnt_d, E_GD);

    // o_gene = sage_gg(h_gene) + sage_dg(h_dis->gene)       (no relu)
    hipLaunchKernelGGL((sage_combine_kernel<true, false>), grid_gene, comb_block, 0, stream,
                       o_gene,
                       agg_g_a, cnt_g_a, W2_gg_l, b2_gg, h_gene, W2_gg_r,
                       agg_g_b, cnt_g_b, W2_dg_l, b2_dg, W2_dg_r);
    // o_dis = sage_gd(h_gene->disease)                      (no relu)
    hipLaunchKernelGGL((sage_combine_kernel<false, false>), grid_dis, comb_block, 0, stream,
                       o_dis,
                       agg_d, cnt_d, W2_gd_l, b2_gd, h_dis, W2_gd_r,
                       nullptr, nullptr, nullptr, nullptr, nullptr);
}